// EnsembleClassifier_66357244723446
// MI455X (gfx1250) — compile-verified
//
#include <hip/hip_runtime.h>
#include <hip/hip_bf16.h>

typedef __attribute__((ext_vector_type(2))) float v2f;
typedef __attribute__((ext_vector_type(8))) float v8f;

#define B_N   2048
#define T_N   2048
#define D_N   32
#define H1    256
#define H2    128
#define M_N   3
#define OUT_N 2
#define EPSF  1e-5f

// ---------------------------------------------------------------------------
// Phase 1: masked mean / unbiased std / last  ->  feats[B, 96]
// One block per batch row. lane = channel d (0..31), wave = t-slice.
// Coalesced: each wave load = x[b, t, 0..31] = 128 contiguous bytes.
// Double accumulators so (sumsq - n*mean^2) rearrangement stays accurate.
// ---------------------------------------------------------------------------
__global__ void feats_kernel(const float* __restrict__ x,
                             const int* __restrict__ lengths,
                             float* __restrict__ feats) {
    int b   = blockIdx.x;
    int tid = threadIdx.x;          // 0..255
    int d   = tid & 31;
    int tg  = tid >> 5;             // 0..7 (wave id)
    int len = lengths[b];
    const float* xb = x + (size_t)b * T_N * D_N;

    double s = 0.0, sq = 0.0;
    for (int t = tg; t < len; t += 8) {
        float v = xb[t * D_N + d];
        s  += (double)v;
        sq += (double)v * (double)v;
    }

    __shared__ double ls[256], lq[256];
    ls[tid] = s;
    lq[tid] = sq;
    __syncthreads();

    if (tid < 32) {
        double ts = 0.0, tq = 0.0;
        for (int w = 0; w < 8; ++w) { ts += ls[d + 32 * w]; tq += lq[d + 32 * w]; }
        double cnt  = (double)len;
        double mean = ts / cnt;
        double var  = (tq - cnt * mean * mean) / (cnt - 1.0);
        if (var < 0.0) var = 0.0;
        float last = xb[(size_t)(len - 1) * D_N + d];
        float* fb = feats + (size_t)b * 96;
        fb[d]      = (float)mean;
        fb[32 + d] = (float)sqrt(var);
        fb[64 + d] = last;
    }
}

// ---------------------------------------------------------------------------
// Layer 1 GEMM: z1[m,b,h] = feats[b,:96] . W1[m,h,:96] + b1[m,h]
// One wave per 16x16 output tile; 24 x V_WMMA_F32_16X16X4_F32 (K=96).
// ---------------------------------------------------------------------------
__global__ void __launch_bounds__(32)
gemm1_kernel(const float* __restrict__ feats, const float* __restrict__ W1,
             const float* __restrict__ b1, float* __restrict__ z1) {
    int lane = threadIdx.x;
    int half = lane >> 4;           // selects K-pair
    int lo   = lane & 15;           // row (A) / col (B)
    int rowBase = blockIdx.x * 16;
    int colBase = blockIdx.y * 16;
    int m       = blockIdx.z;

    const float* Arow = feats + (size_t)(rowBase + lo) * 96 + 2 * half;
    const float* Brow = W1 + (size_t)m * H1 * 96 + (size_t)(colBase + lo) * 96 + 2 * half;

    v8f acc = {};
    for (int k0 = 0; k0 < 96; k0 += 4) {
        v2f a, bm;
        a.x  = Arow[k0];     a.y  = Arow[k0 + 1];
        bm.x = Brow[k0];     bm.y = Brow[k0 + 1];
        acc = __builtin_amdgcn_wmma_f32_16x16x4_f32(
            false, a, false, bm, (short)0, acc, false, false);
    }

    int   col  = colBase + lo;
    float bias = b1[m * H1 + col];
    for (int r = 0; r < 8; ++r) {
        int row = rowBase + r + 8 * half;
        z1[((size_t)m * B_N + row) * H1 + col] = acc[r] + bias;
    }
}

// ---------------------------------------------------------------------------
// BatchNorm statistics (training mode, biased var over batch):
//   scale[m,h] = g / sqrt(var + eps);  shift[m,h] = beta - mean * scale
// One block per flattened (m*F + h) feature; deterministic tree reduction.
// ---------------------------------------------------------------------------
__global__ void bnstats_kernel(const float* __restrict__ z,
                               const float* __restrict__ g,
                               const float* __restrict__ be, int F,
                               float* __restrict__ scale,
                               float* __restrict__ shift) {
    int f   = blockIdx.x;                    // m*F + h
    int tid = threadIdx.x;                   // 256 threads
    const float* col = z + (size_t)(f / F) * B_N * F + (f % F);

    float s = 0.f, sq = 0.f;
    for (int b = tid; b < B_N; b += 256) {
        float v = col[(size_t)b * F];
        s += v; sq += v * v;
    }
    __shared__ float ls[256], lq[256];
    ls[tid] = s; lq[tid] = sq;
    __syncthreads();
    for (int off = 128; off > 0; off >>= 1) {
        if (tid < off) { ls[tid] += ls[tid + off]; lq[tid] += lq[tid + off]; }
        __syncthreads();
    }
    if (tid == 0) {
        float mean = ls[0] * (1.0f / B_N);
        float var  = lq[0] * (1.0f / B_N) - mean * mean;
        if (var < 0.f) var = 0.f;
        float sc = g[f] * rsqrtf(var + EPSF);
        scale[f] = sc;
        shift[f] = be[f] - mean * sc;
    }
}

// ---------------------------------------------------------------------------
// Layer 2 GEMM with fused BN+ReLU on the A operand:
//   z2[m,b,k] = relu(bn(z1[m,b,:])) . W2[m,k,:256] + b2[m,k]
// 64 x V_WMMA_F32_16X16X4_F32 (K=256) per 16x16 tile.
// ---------------------------------------------------------------------------
__global__ void __launch_bounds__(32)
gemm2_kernel(const float* __restrict__ z1, const float* __restrict__ sc1,
             const float* __restrict__ sh1, const float* __restrict__ W2,
             const float* __restrict__ b2, float* __restrict__ z2) {
    int lane = threadIdx.x;
    int half = lane >> 4;
    int lo   = lane & 15;
    int rowBase = blockIdx.x * 16;
    int colBase = blockIdx.y * 16;
    int m       = blockIdx.z;

    const float* Arow = z1 + (size_t)m * B_N * H1 + (size_t)(rowBase + lo) * H1 + 2 * half;
    const float* Brow = W2 + (size_t)m * H2 * H1 + (size_t)(colBase + lo) * H1 + 2 * half;
    const float* sc = sc1 + m * H1;
    const float* sh = sh1 + m * H1;

    v8f acc = {};
    for (int k0 = 0; k0 < H1; k0 += 4) {
        int ka = k0 + 2 * half;
        v2f a, bm;
        a.x  = fmaxf(Arow[k0]     * sc[ka]     + sh[ka],     0.f);
        a.y  = fmaxf(Arow[k0 + 1] * sc[ka + 1] + sh[ka + 1], 0.f);
        bm.x = Brow[k0];  bm.y = Brow[k0 + 1];
        acc = __builtin_amdgcn_wmma_f32_16x16x4_f32(
            false, a, false, bm, (short)0, acc, false, false);
    }

    int   col  = colBase + lo;
    float bias = b2[m * H2 + col];
    for (int r = 0; r < 8; ++r) {
        int row = rowBase + r + 8 * half;
        z2[((size_t)m * B_N + row) * H2 + col] = acc[r] + bias;
    }
}

// ---------------------------------------------------------------------------
// Layer 3 + ensemble mean: out[b,o] = (1/M) * sum_m relu(bn(z2[m,b,:])).W3[m,o,:] + b3
// K=128, OUT=2 — too skinny for WMMA; one thread per batch row.
// ---------------------------------------------------------------------------
__global__ void final_kernel(const float* __restrict__ z2,
                             const float* __restrict__ sc2,
                             const float* __restrict__ sh2,
                             const float* __restrict__ W3,
                             const float* __restrict__ b3,
                             float* __restrict__ out) {
    int b = blockIdx.x * blockDim.x + threadIdx.x;
    if (b >= B_N) return;
    float o0 = 0.f, o1 = 0.f;
    for (int m = 0; m < M_N; ++m) {
        float s0 = b3[m * OUT_N + 0];
        float s1 = b3[m * OUT_N + 1];
        const float* row = z2 + ((size_t)m * B_N + b) * H2;
        const float* sc  = sc2 + m * H2;
        const float* sh  = sh2 + m * H2;
        const float* w0  = W3 + (size_t)(m * OUT_N + 0) * H2;
        const float* w1  = W3 + (size_t)(m * OUT_N + 1) * H2;
        for (int k = 0; k < H2; ++k) {
            float a = fmaxf(row[k] * sc[k] + sh[k], 0.f);
            s0 += a * w0[k];
            s1 += a * w1[k];
        }
        o0 += s0; o1 += s1;
    }
    out[(size_t)b * 2 + 0] = o0 * (1.f / 3.f);
    out[(size_t)b * 2 + 1] = o1 * (1.f / 3.f);
}

// ---------------------------------------------------------------------------
extern "C" void kernel_launch(void* const* d_in, const int* in_sizes, int n_in,
                              void* d_out, int out_size, void* d_ws, size_t ws_size,
                              hipStream_t stream) {
    const float* x       = (const float*)d_in[0];
    const int*   lengths = (const int*)  d_in[1];
    const float* W1      = (const float*)d_in[2];
    const float* b1      = (const float*)d_in[3];
    const float* g1      = (const float*)d_in[4];
    const float* be1     = (const float*)d_in[5];
    const float* W2      = (const float*)d_in[6];
    const float* b2      = (const float*)d_in[7];
    const float* g2      = (const float*)d_in[8];
    const float* be2     = (const float*)d_in[9];
    const float* W3      = (const float*)d_in[10];
    const float* b3      = (const float*)d_in[11];
    float* out = (float*)d_out;

    char* ws = (char*)d_ws;
    // Workspace layout (bytes):
    float* feats = (float*)(ws);                 // 2048*96*4      =   786432
    float* z1    = (float*)(ws + 786432);        // 3*2048*256*4   =  6291456
    float* z2    = (float*)(ws + 7077888);       // 3*2048*128*4   =  3145728
    float* sc1   = (float*)(ws + 10223616);      // 768*4          =     3072
    float* sh1   = (float*)(ws + 10226688);      // 768*4          =     3072
    float* sc2   = (float*)(ws + 10229760);      // 384*4          =     1536
    float* sh2   = (float*)(ws + 10231296);      // 384*4          =     1536

    feats_kernel<<<B_N, 256, 0, stream>>>(x, lengths, feats);

    gemm1_kernel<<<dim3(B_N / 16, H1 / 16, M_N), 32, 0, stream>>>(feats, W1, b1, z1);

    bnstats_kernel<<<M_N * H1, 256, 0, stream>>>(z1, g1, be1, H1, sc1, sh1);

    gemm2_kernel<<<dim3(B_N / 16, H2 / 16, M_N), 32, 0, stream>>>(z1, sc1, sh1, W2, b2, z2);

    bnstats_kernel<<<M_N * H2, 256, 0, stream>>>(z2, g2, be2, H2, sc2, sh2);

    final_kernel<<<(B_N + 255) / 256, 256, 0, stream>>>(z2, sc2, sh2, W3, b3, out);
}